// GridNetDir_39548058862134
// MI455X (gfx1250) — compile-verified
//
#include <hip/hip_runtime.h>
#include <math.h>

typedef float v2f __attribute__((ext_vector_type(2)));
typedef float v4f __attribute__((ext_vector_type(4)));
typedef float v8f __attribute__((ext_vector_type(8)));

#define PI_F 3.14159265358979323846f

__device__ __forceinline__ v8f wmma_f32_16x16x4(v2f a, v2f b, v8f c) {
  // D = A(16x4,f32) * B(4x16,f32) + C(16x16,f32), wave32
  return __builtin_amdgcn_wmma_f32_16x16x4_f32(false, a, false, b, (short)0, c,
                                               false, false);
}

// Block = 128 threads (4 waves). Each wave handles 32 points (two 16-row WMMA
// M-tiles). Layer 1 (16->64) runs on v_wmma_f32_16x16x4_f32; layer 2 (64->3)
// is per-lane VALU.
__global__ __launch_bounds__(128) void GridNetDir_wmma_kernel(
    const float* __restrict__ x, const float* __restrict__ grid,
    const float* __restrict__ w1, const float* __restrict__ b1,
    const float* __restrict__ w2, const float* __restrict__ b2,
    float* __restrict__ out) {
  // padded strides: 20 (=80B, 16B aligned, conflict-free rows),
  //                 68 (=272B, 16B aligned, distinct row banks per half-wave)
  __shared__ float s_w1t[64 * 20];       // w1 transposed: [n][k]
  __shared__ float s_b1[64];
  __shared__ float s_w2[64 * 3];
  __shared__ float s_b2[4];
  __shared__ float s_feat[4][32][20];    // per-wave feat staging
  __shared__ float s_h[4][32][68];       // per-wave hidden staging

  const int tid  = threadIdx.x;
  const int w    = tid >> 5;
  const int lane = tid & 31;
  const int l16  = lane & 15;
  const int hi   = lane >> 4;            // 0: lanes 0-15, 1: lanes 16-31

  // ---- cooperative weight preload ----
  for (int i = tid; i < 1024; i += 128) {       // w1 [16][64] -> w1t [n][k]
    int k = i >> 6, n = i & 63;
    s_w1t[n * 20 + k] = w1[i];
  }
  for (int i = tid; i < 192; i += 128) s_w2[i] = w2[i];
  if (tid < 64) s_b1[tid] = b1[tid];
  if (tid < 3)  s_b2[tid] = b2[tid];
  __syncthreads();

  // ---- phase 1: per-lane quad-linear gather (the bandwidth-bound part) ----
  const int p = blockIdx.x * 128 + tid;
  v4f xv = *(const v4f*)(x + 4 * (size_t)p);
  float c0 = xv.x * (63.0f / PI_F);
  float c1 = (xv.y + PI_F) * (63.0f / (2.0f * PI_F));
  float c2 = (xv.z - 0.5f * PI_F) * (31.0f / (0.35f * PI_F));
  float c3 = (xv.w + 0.85f * PI_F) * (31.0f / (0.35f * PI_F));
  int tlx = (int)c0, tly = (int)c1, tlz = (int)c2, tlw = (int)c3;
  float xf = c0 - (float)tlx, yf = c1 - (float)tly;
  float zf = c2 - (float)tlz, wf = c3 - (float)tlw;
  const int brx = min(tlx + 1, 63), bry = min(tly + 1, 63);
  const int brz = min(tlz + 1, 31), brw = min(tlw + 1, 31);

  // grid element offset (in F=16-float units) = y*65536 + x*1024 + z*32 + wdim
  const int   oy[2] = {tly * 65536, bry * 65536};
  const int   ox[2] = {tlx * 1024,  brx * 1024};
  const int   oz[2] = {tlz * 32,    brz * 32};
  const int   ow[2] = {tlw,         brw};
  const float wy[2] = {1.0f - yf, yf};
  const float wx[2] = {1.0f - xf, xf};
  const float wz[2] = {1.0f - zf, zf};
  const float ww[2] = {1.0f - wf, wf};

  v4f f0 = {0.f, 0.f, 0.f, 0.f}, f1 = f0, f2 = f0, f3 = f0;
#pragma unroll
  for (int iy = 0; iy < 2; ++iy)
#pragma unroll
    for (int ix = 0; ix < 2; ++ix)
#pragma unroll
      for (int iz = 0; iz < 2; ++iz)
#pragma unroll
        for (int iw = 0; iw < 2; ++iw) {
          float wgt = wy[iy] * wx[ix] * wz[iz] * ww[iw];
          const v4f* gp = (const v4f*)grid +
                          (size_t)(oy[iy] + ox[ix] + oz[iz] + ow[iw]) * 4;
          f0 += wgt * gp[0];
          f1 += wgt * gp[1];
          f2 += wgt * gp[2];
          f3 += wgt * gp[3];
        }
  {
    float* fr = &s_feat[w][lane][0];
    *(v4f*)(fr + 0)  = f0;
    *(v4f*)(fr + 4)  = f1;
    *(v4f*)(fr + 8)  = f2;
    *(v4f*)(fr + 12) = f3;
  }
  __syncthreads();

  // ---- phase 2: layer 1 via fp32 WMMA; bias folded into C ----
  v8f acc0[4], acc1[4];
#pragma unroll
  for (int nt = 0; nt < 4; ++nt) {
    float bv = s_b1[16 * nt + l16];        // each lane's N-column bias
    v8f t = {bv, bv, bv, bv, bv, bv, bv, bv};
    acc0[nt] = t;
    acc1[nt] = t;
  }
#pragma unroll
  for (int ks = 0; ks < 4; ++ks) {
    const int ko = 4 * ks + 2 * hi;        // K-pair this half-wave supplies
    v2f a0 = *(const v2f*)&s_feat[w][l16][ko];        // rows 0..15
    v2f a1 = *(const v2f*)&s_feat[w][16 + l16][ko];   // rows 16..31
#pragma unroll
    for (int nt = 0; nt < 4; ++nt) {
      v2f b = *(const v2f*)&s_w1t[(16 * nt + l16) * 20 + ko];
      acc0[nt] = wmma_f32_16x16x4(a0, b, acc0[nt]);
      acc1[nt] = wmma_f32_16x16x4(a1, b, acc1[nt]);
    }
  }

  // ---- phase 3: LeakyReLU + scatter hidden to LDS (C-layout -> row-major) --
#pragma unroll
  for (int nt = 0; nt < 4; ++nt) {
#pragma unroll
    for (int r = 0; r < 8; ++r) {
      float v0 = acc0[nt][r];
      v0 = (v0 >= 0.0f) ? v0 : 0.01f * v0;
      s_h[w][r + 8 * hi][16 * nt + l16] = v0;
      float v1 = acc1[nt][r];
      v1 = (v1 >= 0.0f) ? v1 : 0.01f * v1;
      s_h[w][16 + r + 8 * hi][16 * nt + l16] = v1;
    }
  }
  __syncthreads();

  // ---- phase 4: layer 2 (64 -> 3) + sigmoid, one point per lane ----
  float o0 = s_b2[0], o1 = s_b2[1], o2 = s_b2[2];
  const float* hrow = &s_h[w][lane][0];
#pragma unroll
  for (int k = 0; k < 64; k += 4) {
    v4f hv = *(const v4f*)(hrow + k);
    o0 += hv.x * s_w2[(k + 0) * 3 + 0] + hv.y * s_w2[(k + 1) * 3 + 0] +
          hv.z * s_w2[(k + 2) * 3 + 0] + hv.w * s_w2[(k + 3) * 3 + 0];
    o1 += hv.x * s_w2[(k + 0) * 3 + 1] + hv.y * s_w2[(k + 1) * 3 + 1] +
          hv.z * s_w2[(k + 2) * 3 + 1] + hv.w * s_w2[(k + 3) * 3 + 1];
    o2 += hv.x * s_w2[(k + 0) * 3 + 2] + hv.y * s_w2[(k + 1) * 3 + 2] +
          hv.z * s_w2[(k + 2) * 3 + 2] + hv.w * s_w2[(k + 3) * 3 + 2];
  }
  float* op = out + 3 * (size_t)p;
  op[0] = 255.0f / (1.0f + __expf(-o0));
  op[1] = 255.0f / (1.0f + __expf(-o1));
  op[2] = 255.0f / (1.0f + __expf(-o2));
}

// Scalar tail for N not divisible by 128 (not hit for N = 1,048,576).
__global__ void GridNetDir_tail_kernel(
    const float* __restrict__ x, const float* __restrict__ grid,
    const float* __restrict__ w1, const float* __restrict__ b1,
    const float* __restrict__ w2, const float* __restrict__ b2,
    float* __restrict__ out, int start, int N) {
  int p = start + blockIdx.x * 64 + threadIdx.x;
  if (p >= N) return;
  float xv0 = x[4 * (size_t)p + 0], xv1 = x[4 * (size_t)p + 1];
  float xv2 = x[4 * (size_t)p + 2], xv3 = x[4 * (size_t)p + 3];
  float c0 = xv0 * (63.0f / PI_F);
  float c1 = (xv1 + PI_F) * (63.0f / (2.0f * PI_F));
  float c2 = (xv2 - 0.5f * PI_F) * (31.0f / (0.35f * PI_F));
  float c3 = (xv3 + 0.85f * PI_F) * (31.0f / (0.35f * PI_F));
  int tlx = (int)c0, tly = (int)c1, tlz = (int)c2, tlw = (int)c3;
  float xf = c0 - tlx, yf = c1 - tly, zf = c2 - tlz, wf = c3 - tlw;
  int brx = min(tlx + 1, 63), bry = min(tly + 1, 63);
  int brz = min(tlz + 1, 31), brw = min(tlw + 1, 31);
  const int   oy[2] = {tly * 65536, bry * 65536};
  const int   ox[2] = {tlx * 1024,  brx * 1024};
  const int   oz[2] = {tlz * 32,    brz * 32};
  const int   ow[2] = {tlw,         brw};
  const float wy[2] = {1.0f - yf, yf}, wx[2] = {1.0f - xf, xf};
  const float wz[2] = {1.0f - zf, zf}, ww[2] = {1.0f - wf, wf};
  float feat[16];
#pragma unroll
  for (int f = 0; f < 16; ++f) feat[f] = 0.0f;
  for (int iy = 0; iy < 2; ++iy)
    for (int ix = 0; ix < 2; ++ix)
      for (int iz = 0; iz < 2; ++iz)
        for (int iw = 0; iw < 2; ++iw) {
          float wgt = wy[iy] * wx[ix] * wz[iz] * ww[iw];
          const float* gp =
              grid + (size_t)(oy[iy] + ox[ix] + oz[iz] + ow[iw]) * 16;
#pragma unroll
          for (int f = 0; f < 16; ++f) feat[f] += wgt * gp[f];
        }
  float o[3] = {b2[0], b2[1], b2[2]};
  for (int n = 0; n < 64; ++n) {
    float hn = b1[n];
#pragma unroll
    for (int k = 0; k < 16; ++k) hn += feat[k] * w1[k * 64 + n];
    hn = (hn >= 0.0f) ? hn : 0.01f * hn;
    o[0] += hn * w2[n * 3 + 0];
    o[1] += hn * w2[n * 3 + 1];
    o[2] += hn * w2[n * 3 + 2];
  }
  for (int j = 0; j < 3; ++j)
    out[3 * (size_t)p + j] = 255.0f / (1.0f + __expf(-o[j]));
}

extern "C" void kernel_launch(void* const* d_in, const int* in_sizes, int n_in,
                              void* d_out, int out_size, void* d_ws,
                              size_t ws_size, hipStream_t stream) {
  const float* x    = (const float*)d_in[0];
  const float* grid = (const float*)d_in[1];
  const float* w1   = (const float*)d_in[2];
  const float* b1   = (const float*)d_in[3];
  const float* w2   = (const float*)d_in[4];
  const float* b2   = (const float*)d_in[5];
  float* out = (float*)d_out;

  const int N = in_sizes[0] / 4;      // x is [N,4]
  const int nfull = N / 128;
  if (nfull > 0) {
    GridNetDir_wmma_kernel<<<nfull, 128, 0, stream>>>(x, grid, w1, b1, w2, b2,
                                                      out);
  }
  const int rem = N - nfull * 128;
  if (rem > 0) {
    GridNetDir_tail_kernel<<<(rem + 63) / 64, 64, 0, stream>>>(
        x, grid, w1, b1, w2, b2, out, nfull * 128, N);
  }
}